// MoELayer_85856396247455
// MI455X (gfx1250) — compile-verified
//
#include <hip/hip_runtime.h>
#include <hip/hip_bf16.h>

// ---------------------------------------------------------------------------
// MoE: x[B,D] -> gate(top-2 of E=8) -> per-expert MLP (D->H relu H->O) -> mix
// B=8192 D=1024 E=8 H=4096 O=1024 K=2, all fp32.
// Routed (top-2 only): 4x FLOP cut vs dense reference, identical math.
// Expert kernel: V_WMMA_F32_16X16X4_F32 + TDM tensor_load_to_lds (W1 tiles,
// HW row padding -> conflict-free LDS stride) + global_load_async_to_lds_b128
// (gathered x rows) + global_prefetch on the W2 stream.
// ---------------------------------------------------------------------------

#define MOE_B 8192
#define MOE_D 1024
#define MOE_E 8
#define MOE_H 4096
#define MOE_O 1024

typedef float v2f __attribute__((ext_vector_type(2)));
typedef float v8f __attribute__((ext_vector_type(8)));
typedef unsigned int u32x4 __attribute__((ext_vector_type(4)));
typedef int i32x4 __attribute__((ext_vector_type(4)));
typedef int i32x8 __attribute__((ext_vector_type(8)));

static __device__ __forceinline__ v8f v8zero() {
  v8f v;
#pragma unroll
  for (int i = 0; i < 8; ++i) v[i] = 0.0f;
  return v;
}

// Generic pointers to LDS carry the DS byte offset in their low 32 bits
// (flat-aperture rule).
static __device__ __forceinline__ unsigned lds_off(const void* p) {
  return (unsigned)(unsigned long long)(size_t)p;
}

// ---------------------------------------------------------------------------
// Kernel 0: zero output + routing counters (d_out/d_ws are poisoned by harness)
// ---------------------------------------------------------------------------
__global__ void moe_init(float* __restrict__ out, int n, int* __restrict__ counts) {
  int i = blockIdx.x * blockDim.x + threadIdx.x;
  int stride = gridDim.x * blockDim.x;
  for (; i < n; i += stride) out[i] = 0.0f;
  if (blockIdx.x == 0 && threadIdx.x < MOE_E) counts[threadIdx.x] = 0;
}

// ---------------------------------------------------------------------------
// Kernel 1: gating. One wave32 per token: logits = x@gate_w + gate_b,
// softmax, top-2, renormalize, scatter into per-expert token lists.
// ---------------------------------------------------------------------------
__global__ __launch_bounds__(256) void moe_gate(
    const float* __restrict__ x, const float* __restrict__ gw,
    const float* __restrict__ gb, int* __restrict__ counts,
    int* __restrict__ perm, float* __restrict__ pwgt) {
  const int lane = threadIdx.x & 31;
  const int wave = threadIdx.x >> 5;
  const int t = blockIdx.x * 8 + wave;  // 8 waves per block
  if (t >= MOE_B) return;

  float acc[MOE_E];
#pragma unroll
  for (int e = 0; e < MOE_E; ++e) acc[e] = 0.0f;

  const float* xr = x + (size_t)t * MOE_D;
  for (int d = lane; d < MOE_D; d += 32) {
    const float xv = xr[d];
    const float* g = gw + (size_t)d * MOE_E;
#pragma unroll
    for (int e = 0; e < MOE_E; ++e) acc[e] += xv * g[e];
  }
#pragma unroll
  for (int off = 16; off > 0; off >>= 1) {
#pragma unroll
    for (int e = 0; e < MOE_E; ++e) acc[e] += __shfl_xor(acc[e], off, 32);
  }

  if (lane == 0) {
    float lg[MOE_E];
#pragma unroll
    for (int e = 0; e < MOE_E; ++e) lg[e] = acc[e] + gb[e];
    float m = lg[0];
#pragma unroll
    for (int e = 1; e < MOE_E; ++e) m = lg[e] > m ? lg[e] : m;
    float p[MOE_E];
#pragma unroll
    for (int e = 0; e < MOE_E; ++e) p[e] = expf(lg[e] - m);
    int i1 = 0;
#pragma unroll
    for (int e = 1; e < MOE_E; ++e) if (p[e] > p[i1]) i1 = e;
    int i2 = (i1 == 0) ? 1 : 0;
#pragma unroll
    for (int e = 0; e < MOE_E; ++e) if (e != i1 && p[e] > p[i2]) i2 = e;
    const float inv = 1.0f / (p[i1] + p[i2]);
    const float w1 = p[i1] * inv;
    const float w2 = p[i2] * inv;
    int s1 = atomicAdd(&counts[i1], 1);
    perm[i1 * MOE_B + s1] = t;
    pwgt[i1 * MOE_B + s1] = w1;
    int s2 = atomicAdd(&counts[i2], 1);
    perm[i2 * MOE_B + s2] = t;
    pwgt[i2 * MOE_B + s2] = w2;
  }
}

// ---------------------------------------------------------------------------
// Kernel 2: fused expert MLP on WMMA fp32 with TDM / async-LDS staging.
// Block = 256 threads (8 waves): 16 tokens of expert e, all O=1024 cols.
// ---------------------------------------------------------------------------
#define TILE_M 16
#define HC 128        // H chunk (h tile cols)
#define DC 64         // D chunk (K rows of W1 tile per TDM transfer)
#define XS_STRIDE 68  // (m*68)%64 == 4m%64 -> conflict-free A-fragment reads
#define WL_STRIDE 132 // TDM row pad: 128 DW + 4 DW -> stride 132
#define HS_STRIDE 132

__global__ __launch_bounds__(256) void moe_expert(
    const float* __restrict__ x, const float* __restrict__ W1,
    const float* __restrict__ b1, const float* __restrict__ W2,
    const float* __restrict__ b2, const int* __restrict__ counts,
    const int* __restrict__ perm, const float* __restrict__ pwgt,
    float* __restrict__ out) {
  __shared__ float xs[TILE_M * XS_STRIDE];  //  4.3 KB  x tile 16xDC
  __shared__ float wlds[DC * WL_STRIDE];    // 33.8 KB  W1 tile DCxHC (padded)
  __shared__ float hs[TILE_M * HS_STRIDE];  //  8.4 KB  h tile 16xHC
  __shared__ int tokS[TILE_M];
  __shared__ float wgtS[TILE_M];

  const int e = blockIdx.y;
  const int cnt = counts[e];
  const int row0 = blockIdx.x * TILE_M;
  if (row0 >= cnt) return;  // block-uniform: EXEC all-ones at every WMMA
  const int nrows = (cnt - row0) < TILE_M ? (cnt - row0) : TILE_M;

  const int tid = threadIdx.x;
  const int lane = tid & 31;
  const int wv = tid >> 5;     // 0..7
  const int m16 = lane & 15;   // N (B/C/D) or M (A) index within 16
  const int hi = lane >> 4;    // half-wave select
  const int koff = hi * 2;     // A/B K offset for upper half-wave
  const int moff = hi * 8;     // C/D M offset for upper half-wave

  if (tid < TILE_M) {
    const int r = tid;
    const int idx = (r < nrows) ? (row0 + r) : row0;  // clamp: safe dummy rows
    tokS[r] = perm[e * MOE_B + idx];
    wgtS[r] = (r < nrows) ? pwgt[e * MOE_B + idx] : 0.0f;
  }
  __syncthreads();

  const float* W1e = W1 + (size_t)e * MOE_D * MOE_H;
  const float* W2e = W2 + (size_t)e * MOE_H * MOE_O;

  const unsigned xs_base = lds_off(xs);
  const unsigned wl_base = lds_off(wlds);

  // x-staging coordinates (one async b128 per thread per D-chunk)
  const int sr = tid >> 4;          // row 0..15
  const int sc = (tid & 15) * 4;    // 4-float column group
  const float* xrow = x + (size_t)tokS[sr] * MOE_D + sc;
  const unsigned xs_dst = xs_base + (unsigned)((sr * XS_STRIDE + sc) * 4);

  v8f oacc[8];
#pragma unroll
  for (int j = 0; j < 8; ++j) oacc[j] = v8zero();

  for (int h0 = 0; h0 < MOE_H; h0 += HC) {
    const int hcol = h0 + wv * 16 + m16;
    const float bb = b1[e * MOE_H + hcol];
    v8f hacc;
#pragma unroll
    for (int r = 0; r < 8; ++r) hacc[r] = bb;

    for (int d0 = 0; d0 < MOE_D; d0 += DC) {
      // ---- stage x[16 x DC] via async global->LDS (ASYNCcnt path)
      asm volatile("global_load_async_to_lds_b128 %0, %1, off"
                   :: "v"(xs_dst), "v"((unsigned long long)(size_t)(xrow + d0))
                   : "memory");

      // ---- stage W1[DC x HC] via TDM with HW row padding (TENSORcnt path)
      if (wv == 0) {
        const unsigned long long ga =
            (unsigned long long)(size_t)(W1e + (size_t)d0 * MOE_H + h0);
        u32x4 g0;
        g0[0] = 1u;                                  // count=1, user mode
        g0[1] = wl_base;                             // lds_addr
        g0[2] = (unsigned)ga;                        // global_addr[31:0]
        g0[3] = (unsigned)((ga >> 32) & 0x1FFFFFFu)  // global_addr[56:32]
                | (2u << 30);                        // type=2 (image)
        i32x8 g1;
        g1[0] = (2 << 16)      // data_size = 4 bytes
              | (1 << 20)      // pad_enable
              | (6 << 22)      // pad_interval: every 128 DWORDs
              | (3 << 25);     // pad_amount: 4 DWORDs -> LDS stride 132
        g1[1] = (MOE_H & 0xFFFF) << 16;              // tensor_dim0 lo16
        g1[2] = (MOE_H >> 16) | ((MOE_D & 0xFFFF) << 16);  // dim0 hi | dim1 lo
        g1[3] = (MOE_D >> 16) | (HC << 16);          // dim1 hi | tile_dim0=128
        g1[4] = DC;                                  // tile_dim1=64, tile_dim2=0
        g1[5] = MOE_H;                               // tensor_dim0_stride lo32
        g1[6] = 0;
        g1[7] = 0;
        i32x4 gz;
        gz[0] = 0; gz[1] = 0; gz[2] = 0; gz[3] = 0;
        i32x8 gz8;
#pragma unroll
        for (int q = 0; q < 8; ++q) gz8[q] = 0;
#if __has_builtin(__builtin_amdgcn_tensor_load_to_lds)
        // 6-arg form (this toolchain): (g0, g1, g2, g3, g_extra, cpol)
        __builtin_amdgcn_tensor_load_to_lds(g0, g1, gz, gz, gz8, 0);
#else
        // fallback: cooperative plain staging (wave 0 only reaches here, so
        // do it with all 32 lanes: 64 rows x 128 cols, 2 rows per lane)
        for (int rr = lane; rr < DC; rr += 32) {
          const float* srcw = W1e + (size_t)(d0 + rr) * MOE_H + h0;
          float* dstw = wlds + rr * WL_STRIDE;
          for (int q = 0; q < HC; q += 4)
            *(float4*)(dstw + q) = *(const float4*)(srcw + q);
        }
#endif
      }
      asm volatile("s_wait_asynccnt 0x0" ::: "memory");
#if __has_builtin(__builtin_amdgcn_tensor_load_to_lds)
      __builtin_amdgcn_s_wait_tensorcnt(0);
#endif
      __syncthreads();

      // ---- GEMM1: hacc += xs(16xDC) @ wlds(DCx16 per wave)
#pragma unroll 4
      for (int k0 = 0; k0 < DC; k0 += 4) {
        v2f a = *(const v2f*)(xs + m16 * XS_STRIDE + k0 + koff);
        const int wb = (k0 + koff) * WL_STRIDE + wv * 16 + m16;
        v2f bf;
        bf.x = wlds[wb];               // row k
        bf.y = wlds[wb + WL_STRIDE];   // row k+1  (ds_load_2addr_b32 pair)
        hacc = __builtin_amdgcn_wmma_f32_16x16x4_f32(
            false, a, false, bf, (short)0, hacc, false, false);
      }
      __syncthreads();
    }

    // relu + spill h tile to LDS (C/D layout: VGPR r -> row r+moff, col m16)
#pragma unroll
    for (int r = 0; r < 8; ++r) {
      float v = hacc[r];
      v = v > 0.0f ? v : 0.0f;
      hs[(r + moff) * HS_STRIDE + wv * 16 + m16] = v;
    }
    __syncthreads();

    // ---- GEMM2: oacc += hs(16x128) @ W2[h0:h0+128, 128wv .. +127]
#pragma unroll 2
    for (int k0 = 0; k0 < HC; k0 += 4) {
      v2f a2 = *(const v2f*)(hs + m16 * HS_STRIDE + k0 + koff);
      const float* w2p = W2e + (size_t)(h0 + k0 + koff) * MOE_O + wv * 128 + m16;
      __builtin_prefetch(w2p + 16 * (size_t)MOE_O, 0, 0);  // global_prefetch_b8
#pragma unroll
      for (int j = 0; j < 8; ++j) {
        v2f bf2;
        bf2.x = w2p[j * 16];
        bf2.y = w2p[MOE_O + j * 16];
        oacc[j] = __builtin_amdgcn_wmma_f32_16x16x4_f32(
            false, a2, false, bf2, (short)0, oacc[j], false, false);
      }
    }
    __syncthreads();
  }

  // ---- epilogue: out[token, col] += wgt * (oacc + b2); exactly two
  // commutative fp32 adds per output element on a zeroed buffer.
#pragma unroll
  for (int j = 0; j < 8; ++j) {
    const int ocol = wv * 128 + j * 16 + m16;
    const float bias = b2[e * MOE_O + ocol];
#pragma unroll
    for (int r = 0; r < 8; ++r) {
      const int m = r + moff;
      if (m < nrows) {
        atomicAdd(out + (size_t)tokS[m] * MOE_O + ocol,
                  wgtS[m] * (oacc[j][r] + bias));
      }
    }
  }
}

// ---------------------------------------------------------------------------
// Launch. Workspace: [counts: E ints | perm: E*B ints | pwgt: E*B floats]
// ---------------------------------------------------------------------------
extern "C" void kernel_launch(void* const* d_in, const int* in_sizes, int n_in,
                              void* d_out, int out_size, void* d_ws, size_t ws_size,
                              hipStream_t stream) {
  const float* x = (const float*)d_in[0];
  const float* W1 = (const float*)d_in[1];
  const float* b1 = (const float*)d_in[2];
  const float* W2 = (const float*)d_in[3];
  const float* b2 = (const float*)d_in[4];
  const float* gw = (const float*)d_in[5];
  const float* gb = (const float*)d_in[6];
  float* out = (float*)d_out;

  int* counts = (int*)d_ws;
  int* perm = (int*)((char*)d_ws + 256);
  float* pwgt = (float*)((char*)d_ws + 256 + (size_t)MOE_E * MOE_B * sizeof(int));

  moe_init<<<2048, 256, 0, stream>>>(out, MOE_B * MOE_O, counts);
  moe_gate<<<MOE_B / 8, 256, 0, stream>>>(x, gw, gb, counts, perm, pwgt);
  dim3 grid(MOE_B / TILE_M, MOE_E);
  moe_expert<<<grid, 256, 0, stream>>>(x, W1, b1, W2, b2, counts, perm, pwgt, out);
}